// FedCHG_GAE_34815004902081
// MI455X (gfx1250) — compile-verified
//
#include <hip/hip_runtime.h>
#include <math.h>

typedef __attribute__((ext_vector_type(2))) float v2f;
typedef __attribute__((ext_vector_type(8))) float v8f;

#define TPB 256
#define WPAD 66  // transposed-W LDS row stride: even (b64-aligned) + conflict-breaking

// ---------------------------------------------------------------- precompute
__global__ __launch_bounds__(TPB) void k_init_deg(float* deg, int n) {
  int i = blockIdx.x * TPB + threadIdx.x;
  if (i < n) deg[i] = 1.0f;  // self-loop
}

__global__ __launch_bounds__(TPB) void k_accum_deg(const int* __restrict__ dst,
                                                   float* deg, int e) {
  int i = blockIdx.x * TPB + threadIdx.x;
  if (i < e) atomicAdd(&deg[dst[i]], 1.0f);
}

__global__ __launch_bounds__(TPB) void k_dinv(float* deg, int n) {
  int i = blockIdx.x * TPB + threadIdx.x;
  if (i < n) deg[i] = rsqrtf(deg[i]);
}

__global__ __launch_bounds__(TPB) void k_norm(const int* __restrict__ src,
                                              const int* __restrict__ dst,
                                              const float* __restrict__ dinv,
                                              float* __restrict__ norm, int e) {
  int i = blockIdx.x * TPB + threadIdx.x;
  if (i < e) norm[i] = dinv[src[i]] * dinv[dst[i]];
}

// ------------------------------------------------------------------- encoder
// h = tanh(x @ W_in + b)    x:[N,7]  W:[7,64]
__global__ __launch_bounds__(TPB) void k_enc_lin(const float* __restrict__ x,
                                                 const float* __restrict__ W,
                                                 const float* __restrict__ b,
                                                 float* __restrict__ out, int n) {
  int idx = blockIdx.x * TPB + threadIdx.x;
  if (idx >= n * 64) return;
  int node = idx >> 6, o = idx & 63;
  const float* xr = x + (size_t)node * 7;
  float acc = b[o];
#pragma unroll
  for (int k = 0; k < 7; ++k) acc += xr[k] * W[k * 64 + o];
  out[idx] = tanhf(acc);
}

// ------------------------------------------------------- WMMA dense 64x64 mm
// T = H @ W (+ bias)   H:[N,64]  W:[64,64] fp32, via V_WMMA_F32_16X16X4_F32.
// One wave computes a 16x64 strip: 4 accumulators, 16 K-steps of K=4.
// W is staged TRANSPOSED in LDS so each B fragment (two adjacent K values of
// one column) is a single aligned ds_load_b64 into a contiguous VGPR pair.
__global__ __launch_bounds__(TPB) void k_wmma_mm64(const float* __restrict__ H,
                                                   const float* __restrict__ W,
                                                   const float* __restrict__ bias,
                                                   float* __restrict__ T, int n) {
  __shared__ float sWt[64 * WPAD];  // sWt[col*WPAD + k] = W[k*64 + col]
  for (int i = threadIdx.x; i < 64 * 64; i += TPB) {
    int k = i >> 6, col = i & 63;
    sWt[col * WPAD + k] = W[i];
  }
  __syncthreads();

  int wave = threadIdx.x >> 5;
  int lane = threadIdx.x & 31;
  int row0 = (blockIdx.x * 8 + wave) * 16;
  if (row0 >= n) return;  // wave-uniform

  int l15   = lane & 15;
  int lhalf = lane >> 4;

  // A fragment: lanes 0-15 hold K={k0,k0+1}, lanes 16-31 K={k0+2,k0+3}
  int arow = row0 + l15;
  if (arow >= n) arow = n - 1;  // clamp -> EXEC stays full for WMMA
  const float* Ar = H + (size_t)arow * 64;

  v8f acc[4] = {};
#pragma unroll
  for (int k0 = 0; k0 < 64; k0 += 4) {
    int kb = k0 + lhalf * 2;  // even
    v2f a = *(const v2f*)(Ar + kb);  // global_load_b64
#pragma unroll
    for (int nt = 0; nt < 4; ++nt) {
      int col = nt * 16 + l15;
      v2f bfrag = *(const v2f*)(&sWt[col * WPAD + kb]);  // ds_load_b64
      acc[nt] = __builtin_amdgcn_wmma_f32_16x16x4_f32(
          false, a, false, bfrag, (short)0, acc[nt], false, false);
    }
  }

  // C/D layout: VGPR v -> row row0 + v + (lane>=16 ? 8 : 0), col = nt*16 + l15
#pragma unroll
  for (int nt = 0; nt < 4; ++nt) {
    int col = nt * 16 + l15;
    float bv = bias ? bias[col] : 0.0f;
#pragma unroll
    for (int v = 0; v < 8; ++v) {
      int row = row0 + v + lhalf * 8;
      if (row < n) T[(size_t)row * 64 + col] = acc[nt][v] + bv;
    }
  }
}

// ------------------------------------------------------------ GCN aggregation
// g = b + dinv^2 * t   (self-loop term + bias)
__global__ __launch_bounds__(TPB) void k_agg_init(const float* __restrict__ t,
                                                  const float* __restrict__ dinv,
                                                  const float* __restrict__ b,
                                                  float* __restrict__ g, int n) {
  int idx = blockIdx.x * TPB + threadIdx.x;
  if (idx >= n * 64) return;
  int node = idx >> 6, c = idx & 63;
  float di = dinv[node];
  g[idx] = b[c] + di * di * t[idx];
}

// g[dst] += norm[e] * t[src]   edge-parallel, float4 per thread, L2 atomics
__global__ __launch_bounds__(TPB) void k_agg_scatter(const float* __restrict__ t,
                                                     const int* __restrict__ src,
                                                     const int* __restrict__ dst,
                                                     const float* __restrict__ norm,
                                                     float* g, int e) {
  int idx = blockIdx.x * TPB + threadIdx.x;
  if (idx >= e * 16) return;
  int ed = idx >> 4, q = idx & 15;
  float nv = norm[ed];
  const float4* tv = (const float4*)(t + (size_t)src[ed] * 64);
  float4 v = tv[q];
  float* go = g + (size_t)dst[ed] * 64 + q * 4;
  atomicAdd(go + 0, nv * v.x);
  atomicAdd(go + 1, nv * v.y);
  atomicAdd(go + 2, nv * v.z);
  atomicAdd(go + 3, nv * v.w);
}

__global__ __launch_bounds__(TPB) void k_tanh(const float* __restrict__ in,
                                              float* __restrict__ out, int total) {
  int idx = blockIdx.x * TPB + threadIdx.x;
  if (idx < total) out[idx] = tanhf(in[idx]);
}

// -------------------------------------------------------- 7-wide layers (d3)
__global__ __launch_bounds__(TPB) void k_mm64x7(const float* __restrict__ h,
                                                const float* __restrict__ W,
                                                float* __restrict__ t7, int n) {
  int idx = blockIdx.x * TPB + threadIdx.x;
  if (idx >= n * 7) return;
  int node = idx / 7, o = idx - node * 7;
  const float* hr = h + (size_t)node * 64;
  float acc = 0.0f;
#pragma unroll
  for (int k = 0; k < 64; ++k) acc += hr[k] * W[k * 7 + o];
  t7[idx] = acc;
}

__global__ __launch_bounds__(TPB) void k_agg7_init(const float* __restrict__ t7,
                                                   const float* __restrict__ dinv,
                                                   const float* __restrict__ b,
                                                   float* __restrict__ g, int n) {
  int idx = blockIdx.x * TPB + threadIdx.x;
  if (idx >= n * 7) return;
  int node = idx / 7, o = idx - node * 7;
  float di = dinv[node];
  g[idx] = b[o] + di * di * t7[idx];
}

__global__ __launch_bounds__(TPB) void k_agg7_scatter(const float* __restrict__ t7,
                                                      const int* __restrict__ src,
                                                      const int* __restrict__ dst,
                                                      const float* __restrict__ norm,
                                                      float* g, int e) {
  int idx = blockIdx.x * TPB + threadIdx.x;
  if (idx >= e * 7) return;
  int ed = idx / 7, c = idx - ed * 7;
  atomicAdd(&g[(size_t)dst[ed] * 7 + c], norm[ed] * t7[(size_t)src[ed] * 7 + c]);
}

// recon = hp @ Wd (+b)   hp:[N,7]  Wd:[7,7]
__global__ __launch_bounds__(TPB) void k_dec_post(const float* __restrict__ hp,
                                                  const float* __restrict__ W,
                                                  const float* __restrict__ b,
                                                  float* __restrict__ recon, int n) {
  int idx = blockIdx.x * TPB + threadIdx.x;
  if (idx >= n * 7) return;
  int node = idx / 7, o = idx - node * 7;
  const float* hr = hp + (size_t)node * 7;
  float acc = b[o];
#pragma unroll
  for (int k = 0; k < 7; ++k) acc += hr[k] * W[k * 7 + o];
  recon[idx] = acc;
}

// ------------------------------------------------------------------- launch
static inline dim3 g1(long long t) { return dim3((unsigned)((t + TPB - 1) / TPB)); }

extern "C" void kernel_launch(void* const* d_in, const int* in_sizes, int n_in,
                              void* d_out, int out_size, void* d_ws, size_t ws_size,
                              hipStream_t stream) {
  const float* x          = (const float*)d_in[0];
  const int*   ei         = (const int*)d_in[1];
  const float* enc_lin_W  = (const float*)d_in[2];
  const float* enc_lin_b  = (const float*)d_in[3];
  const float* c1W = (const float*)d_in[4],  *c1b = (const float*)d_in[5];
  const float* c2W = (const float*)d_in[6],  *c2b = (const float*)d_in[7];
  const float* c3W = (const float*)d_in[8],  *c3b = (const float*)d_in[9];
  const float* enc_post_W = (const float*)d_in[10], *enc_post_b = (const float*)d_in[11];
  const float* d1W = (const float*)d_in[12], *d1b = (const float*)d_in[13];
  const float* d2W = (const float*)d_in[14], *d2b = (const float*)d_in[15];
  const float* d3W = (const float*)d_in[16], *d3b = (const float*)d_in[17];
  const float* dec_post_W = (const float*)d_in[18], *dec_post_b = (const float*)d_in[19];

  const int N = in_sizes[0] / 7;
  const int E = in_sizes[1] / 2;
  const int* src = ei;
  const int* dst = ei + E;

  // workspace layout (fp32)
  float* ws   = (float*)d_ws;
  float* dinv = ws;                        // N      (deg -> dinv in place)
  float* nrm  = dinv + N;                  // E
  float* bufA = nrm + E;                   // N*64
  float* bufB = bufA + (size_t)N * 64;     // N*64
  float* bufC = bufB + (size_t)N * 64;     // N*64

  // outputs: (recon, z, h_L, h_prime_L) flat in return order
  float* out_recon = (float*)d_out;                  // N*7
  float* out_z     = out_recon + (size_t)N * 7;      // N*64
  float* out_hL    = out_z + (size_t)N * 64;         // N*64
  float* out_hpL   = out_hL + (size_t)N * 64;        // N*7

  dim3 B(TPB);
  const long long NH = (long long)N * 64;
  const int mmBlocks = (N + 127) / 128;  // 8 waves/block * 16 rows/wave

  // normalization coefficients
  k_init_deg <<<g1(N), B, 0, stream>>>(dinv, N);
  k_accum_deg<<<g1(E), B, 0, stream>>>(dst, dinv, E);
  k_dinv     <<<g1(N), B, 0, stream>>>(dinv, N);
  k_norm     <<<g1(E), B, 0, stream>>>(src, dst, dinv, nrm, E);

  // encoder input lift
  k_enc_lin<<<g1(NH), B, 0, stream>>>(x, enc_lin_W, enc_lin_b, bufA, N);

  // one GCN layer: t = h@W ; g = b + dinv^2*t ; g[dst] += norm*t[src] ; tanh
  auto gcn = [&](const float* hin, const float* W, const float* bb,
                 float* t, float* g, float* act_out) {
    k_wmma_mm64  <<<mmBlocks, B, 0, stream>>>(hin, W, nullptr, t, N);
    k_agg_init   <<<g1(NH), B, 0, stream>>>(t, dinv, bb, g, N);
    k_agg_scatter<<<g1((long long)E * 16), B, 0, stream>>>(t, src, dst, nrm, g, E);
    k_tanh       <<<g1(NH), B, 0, stream>>>(g, act_out, (int)NH);
  };

  gcn(bufA, c1W, c1b, bufB, bufC, bufC);    // h = bufC
  gcn(bufC, c2W, c2b, bufB, bufA, bufA);    // h = bufA
  gcn(bufA, c3W, c3b, bufB, bufC, out_hL);  // h_L -> d_out

  // z = h_L @ enc_post_W + b  (bias fused in GEMM, no aggregation/tanh)
  k_wmma_mm64<<<mmBlocks, B, 0, stream>>>(out_hL, enc_post_W, enc_post_b, out_z, N);

  gcn(out_z, d1W, d1b, bufB, bufA, bufA);   // h = bufA
  gcn(bufA,  d2W, d2b, bufB, bufC, bufC);   // h = bufC

  // d3: gcn(h, 64->7), no tanh -> h_prime_L in d_out
  k_mm64x7      <<<g1((long long)N * 7), B, 0, stream>>>(bufC, d3W, bufB, N);
  k_agg7_init   <<<g1((long long)N * 7), B, 0, stream>>>(bufB, dinv, d3b, out_hpL, N);
  k_agg7_scatter<<<g1((long long)E * 7), B, 0, stream>>>(bufB, src, dst, nrm, out_hpL, E);

  // recon = h_prime_L @ dec_post_W + b
  k_dec_post<<<g1((long long)N * 7), B, 0, stream>>>(out_hpL, dec_post_W, dec_post_b,
                                                     out_recon, N);
}